// SphereTracingRenderer_33913061769933
// MI455X (gfx1250) — compile-verified
//
#include <hip/hip_runtime.h>

typedef __attribute__((ext_vector_type(2))) float v2f;
typedef __attribute__((ext_vector_type(8))) float v8f;

#define NRAYS_BLOCK 256
#define HIDDEN 256
#define NEAR_T 1.0f
#define FAR_T 10.0f
#define MAX_ITERS 64

__global__ __launch_bounds__(NRAYS_BLOCK) void sphere_trace_mlp_kernel(
    const float* __restrict__ origins,
    const float* __restrict__ dirs,
    const float* __restrict__ center,
    const float* __restrict__ radius,
    const float* __restrict__ w1,   // [3][256] row-major
    const float* __restrict__ b1,   // [256]
    const float* __restrict__ w2,   // [256][3] row-major
    const float* __restrict__ b2,   // [3]
    float* __restrict__ out)        // [N][3]
{
    // w1/b1 pre-swizzled into WMMA B-operand (4x16 f32) layout per 16-col tile:
    // lane<16 -> (w1[0][n], w1[1][n]); lane>=16 -> (w1[2][n], b1[n]),  n = 16*j + (lane&15)
    __shared__ v2f   lds_b1t[16 * 32];
    __shared__ float lds_w2[HIDDEN * 3];

    const int tid  = threadIdx.x;
    const int lane = tid & 31;
    const int wave = tid >> 5;
    const int ray  = blockIdx.x * NRAYS_BLOCK + tid;

    // ---- stage weights into LDS ----
    for (int e = tid; e < 16 * 32; e += NRAYS_BLOCK) {
        int j = e >> 5, l = e & 31;
        int n = (j << 4) + (l & 15);
        v2f v;
        if (l < 16) { v.x = w1[0 * HIDDEN + n]; v.y = w1[1 * HIDDEN + n]; }
        else        { v.x = w1[2 * HIDDEN + n]; v.y = b1[n]; }
        lds_b1t[e] = v;
    }
    for (int e = tid; e < HIDDEN * 3; e += NRAYS_BLOCK) lds_w2[e] = w2[e];
    __syncthreads();

    // ---- per-ray sphere tracing (uniform 64 iterations, see analysis) ----
    const float ox = origins[ray * 3 + 0], oy = origins[ray * 3 + 1], oz = origins[ray * 3 + 2];
    const float rdx = dirs[ray * 3 + 0],   rdy = dirs[ray * 3 + 1],   rdz = dirs[ray * 3 + 2];
    const float cx = center[0], cy = center[1], cz = center[2];
    const float R  = radius[0];

    float t  = NEAR_T;
    float px = ox + t * rdx, py = oy + t * rdy, pz = oz + t * rdz;
    bool mask = true;
#pragma unroll 8
    for (int it = 0; it < MAX_ITERS; ++it) {
        float qx = px - cx, qy = py - cy, qz = pz - cz;
        float f = sqrtf(qx * qx + qy * qy + qz * qz) - R;
        t += f;
        px = ox + t * rdx; py = oy + t * rdy; pz = oz + t * rdz;
        mask = mask && (t <= FAR_T);
    }

    const unsigned long long bal = __ballot(mask);   // wave32: bits [31:0]

    // ---- fused MLP: layer1 on f32 WMMA, layer2 per-lane MAC + butterfly ----
    const float b2x = b2[0], b2y = b2[1], b2z = b2[2];
    const int  halfSel = (lane >= 16) ? 1 : 0;
    const int  nlane   = lane & 15;
    const int  waveRay0 = blockIdx.x * NRAYS_BLOCK + (wave << 5);

    for (int pass = 0; pass < 2; ++pass) {
        // A operand (16x4 f32): rows = 16 rays of this pass, cols = (x,y,z,1)
        const int srcLane = nlane + (pass << 4);
        float sx = __shfl(px, srcLane, 32);
        float sy = __shfl(py, srcLane, 32);
        float sz = __shfl(pz, srcLane, 32);
        v2f a;
        a.x = halfSel ? sz : sx;      // lanes 0-15: K=0 (x), lanes 16-31: K=2 (z)
        a.y = halfSel ? 1.0f : sy;    // lanes 0-15: K=1 (y), lanes 16-31: K=3 (1 -> bias row)

        float accR[8], accG[8], accB[8];
#pragma unroll
        for (int v = 0; v < 8; ++v) { accR[v] = 0.f; accG[v] = 0.f; accB[v] = 0.f; }

#pragma unroll
        for (int j = 0; j < 16; ++j) {
            v2f b = lds_b1t[(j << 5) + lane];
            v8f c = {};
            // D = A x B (+0): 16x16 f32 tile of hidden activations (pre-relu)
            v8f dmat = __builtin_amdgcn_wmma_f32_16x16x4_f32(
                false, a, false, b, (short)0, c, false, false);

            const int row = (j << 4) + nlane;           // hidden column index for this lane
            const float wr = lds_w2[row * 3 + 0];
            const float wg = lds_w2[row * 3 + 1];
            const float wb = lds_w2[row * 3 + 2];
#pragma unroll
            for (int v = 0; v < 8; ++v) {
                float h = fmaxf(dmat[v], 0.0f);          // relu; h[m][n], m = v + 8*halfSel
                accR[v] += h * wr;
                accG[v] += h * wg;
                accB[v] += h * wb;
            }
        }

        // reduce across the 16 columns held by each 16-lane half (xor<16 stays in-half)
#pragma unroll
        for (int off = 8; off >= 1; off >>= 1) {
#pragma unroll
            for (int v = 0; v < 8; ++v) {
                accR[v] += __shfl_xor(accR[v], off, 32);
                accG[v] += __shfl_xor(accG[v], off, 32);
                accB[v] += __shfl_xor(accB[v], off, 32);
            }
        }

        const int passBase = waveRay0 + (pass << 4);
#pragma unroll
        for (int v = 0; v < 8; ++v) {
            if (nlane == v) {                            // 2 writer lanes: v and v+16
                const int m      = v + (halfSel << 3);   // ray row within the 16-ray tile
                const int wlLane = (pass << 4) + m;      // wave-local lane owning this ray
                const int r      = passBase + m;
                float s0 = 0.f, s1 = 0.f, s2 = 0.f;
                if ((bal >> wlLane) & 1ull) {
                    s0 = 1.0f / (1.0f + expf(-(accR[v] + b2x)));
                    s1 = 1.0f / (1.0f + expf(-(accG[v] + b2y)));
                    s2 = 1.0f / (1.0f + expf(-(accB[v] + b2z)));
                }
                out[r * 3 + 0] = s0;
                out[r * 3 + 1] = s1;
                out[r * 3 + 2] = s2;
            }
        }
    }
}

extern "C" void kernel_launch(void* const* d_in, const int* in_sizes, int n_in,
                              void* d_out, int out_size, void* d_ws, size_t ws_size,
                              hipStream_t stream) {
    const float* origins = (const float*)d_in[0];
    const float* dirs    = (const float*)d_in[1];
    const float* center  = (const float*)d_in[2];
    const float* radius  = (const float*)d_in[3];
    const float* w1      = (const float*)d_in[4];
    const float* b1      = (const float*)d_in[5];
    const float* w2      = (const float*)d_in[6];
    const float* b2      = (const float*)d_in[7];
    float* out = (float*)d_out;

    const int n_rays = in_sizes[0] / 3;           // 262144
    const int blocks = n_rays / NRAYS_BLOCK;      // exact multiple (1024)

    sphere_trace_mlp_kernel<<<blocks, NRAYS_BLOCK, 0, stream>>>(
        origins, dirs, center, radius, w1, b1, w2, b2, out);
}